// DotProductAttention_27384711479549
// MI455X (gfx1250) — compile-verified
//
#include <hip/hip_runtime.h>

typedef __attribute__((ext_vector_type(2))) float v2f;
typedef __attribute__((ext_vector_type(8))) float v8f;

#define B_    64
#define KV_   8192
#define D_    128
#define SPLIT 8
#define KV_PER (KV_ / SPLIT)          // 1024 rows per block
#define WAVES  8
#define ROWS_PER_WAVE (KV_PER / WAVES) // 128
#define NEG_  (-1000000.0f)
#define QSCALE 0.08838834764831845f    // 1/sqrt(128)
#define WSSTRIDE (D_ + 2)              // [m, l, out[128]] per partition

// ---------------------------------------------------------------------------
// Kernel 1: one block = (batch b, split s). Computes masked scores for its
// 1024 KV rows via V_WMMA_F32_16X16X4_F32, local softmax stats, and the
// partial weighted V-sum, again via WMMA. Writes (m_loc, l_loc, out[128]).
// ---------------------------------------------------------------------------
__global__ __launch_bounds__(256)
void attn_partial(const float* __restrict__ q, const float* __restrict__ k,
                  const float* __restrict__ v, const int* __restrict__ vlen,
                  float* __restrict__ ws)
{
    __shared__ float q_s[D_];
    __shared__ float p_s[KV_PER];
    __shared__ float red_s[256];

    const int s    = blockIdx.x;
    const int b    = blockIdx.y;
    const int tid  = threadIdx.x;
    const int lane = tid & 31;
    const int wave = tid >> 5;
    const int h    = lane >> 4;   // 0: lanes 0-15, 1: lanes 16-31
    const int l15  = lane & 15;
    const int kbase = s * KV_PER;
    const int vl   = vlen[b];

    // load + pre-scale q (folds the 1/sqrt(d) into the scores GEMV)
    if (tid < D_) q_s[tid] = q[b * D_ + tid] * QSCALE;
    __syncthreads();

    // ---------------- pass 1: scores = (q/sqrt(d)) . K^T -------------------
    // A-frag (16x4 f32): lane pair (l, l+16) covers K-row kbase+krow+l15,
    // dims {4c+2h, 4c+2h+1}. B-frag (4x16): q chunk replicated across cols.
    const float* kb = k + (size_t)b * KV_ * D_;
    for (int t = 0; t < ROWS_PER_WAVE / 16; ++t) {
        const int krow = wave * ROWS_PER_WAVE + t * 16;   // block-local row
        const float* krp = kb + (size_t)(kbase + krow + l15) * D_ + 2 * h;
        v8f acc = {0.f, 0.f, 0.f, 0.f, 0.f, 0.f, 0.f, 0.f};
        #pragma unroll
        for (int c = 0; c < 32; ++c) {
            v2f a = *(const v2f*)(krp + 4 * c);           // global b64 / lane
            v2f bq;
            bq.x = q_s[4 * c + h];                        // B row K=0/1
            bq.y = q_s[4 * c + 2 + h];                    // B row K=2/3
            acc = __builtin_amdgcn_wmma_f32_16x16x4_f32(
                    false, a, false, bq, (short)0, acc, false, false);
        }
        // C layout: VGPR r holds M = 8h + r; all 16 columns identical.
        if (l15 == 0) {
            #pragma unroll
            for (int r = 0; r < 8; ++r) {
                const int m = 8 * h + r;
                const int kvIdx = kbase + krow + m;
                p_s[krow + m] = (kvIdx < vl) ? acc[r] : NEG_;
            }
        }
    }
    __syncthreads();

    // ---------------- local max over 1024 scores ---------------------------
    float m4 = fmaxf(fmaxf(p_s[tid], p_s[tid + 256]),
                     fmaxf(p_s[tid + 512], p_s[tid + 768]));
    red_s[tid] = m4;
    __syncthreads();
    for (int off = 128; off > 0; off >>= 1) {
        if (tid < off) red_s[tid] = fmaxf(red_s[tid], red_s[tid + off]);
        __syncthreads();
    }
    const float mloc = red_s[0];
    __syncthreads();

    // ---------------- p = exp(s - mloc), local sum -------------------------
    float sum4 = 0.f;
    #pragma unroll
    for (int i = 0; i < 4; ++i) {
        const float e = expf(p_s[tid + 256 * i] - mloc);
        p_s[tid + 256 * i] = e;      // exclusive indices per thread: no race
        sum4 += e;
    }
    red_s[tid] = sum4;
    __syncthreads();
    for (int off = 128; off > 0; off >>= 1) {
        if (tid < off) red_s[tid] += red_s[tid + off];
        __syncthreads();
    }
    const float lsum = red_s[0];
    __syncthreads();

    // ---------------- pass 2: out_partial = p . V --------------------------
    // Each wave owns output dims [wave*16, wave*16+16). A-frag: p[k0..k0+3]
    // replicated over rows; B-frag: V 4x16 tile (lanes 0-15 row k0+?, etc.)
    const float* vb = v + ((size_t)b * KV_ + kbase) * D_ + wave * 16 + l15;
    v8f o = {0.f, 0.f, 0.f, 0.f, 0.f, 0.f, 0.f, 0.f};
    #pragma unroll 8
    for (int k0 = 0; k0 < KV_PER; k0 += 4) {
        v2f a = *(const v2f*)(p_s + k0 + 2 * h);          // {p[k0+2h], p[k0+2h+1]}
        v2f bv;
        bv.x = vb[(size_t)(k0 + h) * D_];                 // V[k0+h ][col]
        bv.y = vb[(size_t)(k0 + 2 + h) * D_];             // V[k0+2+h][col]
        o = __builtin_amdgcn_wmma_f32_16x16x4_f32(
                false, a, false, bv, (short)0, o, false, false);
    }

    float* wp = ws + (size_t)(b * SPLIT + s) * WSSTRIDE;
    if (tid == 0) { wp[0] = mloc; wp[1] = lsum; }
    // all C rows identical (A rows replicated): lanes 0-15 write their column
    if (h == 0) wp[2 + wave * 16 + l15] = o[0];
}

// ---------------------------------------------------------------------------
// Kernel 2: per-batch log-sum-exp merge of the 8 partitions.
// ---------------------------------------------------------------------------
__global__ __launch_bounds__(128)
void attn_reduce(const float* __restrict__ ws, float* __restrict__ out)
{
    const int b = blockIdx.x;
    const int d = threadIdx.x;
    const float* wp = ws + (size_t)b * SPLIT * WSSTRIDE;

    float gm = -INFINITY;
    #pragma unroll
    for (int s = 0; s < SPLIT; ++s) gm = fmaxf(gm, wp[s * WSSTRIDE]);

    float denom = 0.f, acc = 0.f;
    #pragma unroll
    for (int s = 0; s < SPLIT; ++s) {
        const float w = expf(wp[s * WSSTRIDE] - gm);  // fully-masked part -> 0
        denom += w * wp[s * WSSTRIDE + 1];
        acc   += w * wp[s * WSSTRIDE + 2 + d];
    }
    out[(size_t)b * D_ + d] = acc / denom;
}

// ---------------------------------------------------------------------------
extern "C" void kernel_launch(void* const* d_in, const int* in_sizes, int n_in,
                              void* d_out, int out_size, void* d_ws, size_t ws_size,
                              hipStream_t stream)
{
    const float* q    = (const float*)d_in[0];   // (64, 1, 128)
    const float* k    = (const float*)d_in[1];   // (64, 8192, 128)
    const float* v    = (const float*)d_in[2];   // (64, 8192, 128)
    const int*   vlen = (const int*)d_in[3];     // (64,)
    float* out = (float*)d_out;                  // (64, 1, 128)
    float* ws  = (float*)d_ws;                   // 64*8*130 floats = 260 KB

    attn_partial<<<dim3(SPLIT, B_), 256, 0, stream>>>(q, k, v, vlen, ws);
    attn_reduce<<<B_, 128, 0, stream>>>(ws, out);
}